// Model_48017734369318
// MI455X (gfx1250) — compile-verified
//
#include <hip/hip_runtime.h>
#include <math.h>

#define B_  16
#define C_  256
#define H_  32
#define W_  32
#define HW_ 1024
#define NH_ 16
#define DH_ 16
#define NS_ 256

typedef float v2f __attribute__((ext_vector_type(2)));
typedef float v8f __attribute__((ext_vector_type(8)));

__device__ __forceinline__ v8f wmma4(v2f a, v2f b, v8f c) {
    // D = A(16x4 f32) * B(4x16 f32) + C(16x16 f32)
    return __builtin_amdgcn_wmma_f32_16x16x4_f32(
        false, a, false, b, (short)0, c, false, false);
}

// ---------------------------------------------------------------------------
// Generic fp32 WMMA GEMM: Y[b] = Wm(MxK) * X[b](KxN) + bias, optional BN.
// grid = (B, M/16, N/256); block = 128 (4 waves); wave -> 16x64 strip.
// Software-pipelined: next A/B fragments are in flight while current WMMAs run.
// ---------------------------------------------------------------------------
__global__ __launch_bounds__(128) void gemm_bias_kernel(
    const float* __restrict__ Wm, const float* __restrict__ X,
    const float* __restrict__ bias, float* __restrict__ Y,
    int N, int K, long strideX, long strideY,
    const float* __restrict__ bng, const float* __restrict__ bnb,
    const float* __restrict__ bnm, const float* __restrict__ bnv, int do_bn)
{
    const int lane  = threadIdx.x & 31;
    const int wid   = threadIdx.x >> 5;
    const int li    = lane & 15;
    const int hi    = lane >> 4;
    const int mbase = blockIdx.y * 16;
    const int nb0   = blockIdx.z * 256 + wid * 64;

    const float* Xb = X + (long)blockIdx.x * strideX;
    float*       Yb = Y + (long)blockIdx.x * strideY;
    const float* Wp = Wm + (long)(mbase + li) * K + hi * 2;

    v8f acc[4];
#pragma unroll
    for (int t = 0; t < 4; ++t) acc[t] = (v8f){0.f,0.f,0.f,0.f,0.f,0.f,0.f,0.f};

    const long stepX = 4 * (long)N;
    const float* Xk = Xb + (long)(hi * 2) * N + nb0 + li;

    // prologue: fragments for k = 0
    v2f a = *(const v2f*)(Wp);
    v2f bf[4];
#pragma unroll
    for (int t = 0; t < 4; ++t) { bf[t].x = Xk[t * 16]; bf[t].y = Xk[N + t * 16]; }

    for (int k = 0; k < K - 4; k += 4) {
        const float* Xn = Xk + stepX;
        v2f an = *(const v2f*)(Wp + k + 4);
        v2f bn[4];
#pragma unroll
        for (int t = 0; t < 4; ++t) { bn[t].x = Xn[t * 16]; bn[t].y = Xn[N + t * 16]; }
        __builtin_prefetch(Xn + stepX, 0, 1);          // global_prefetch_b8
#pragma unroll
        for (int t = 0; t < 4; ++t) acc[t] = wmma4(a, bf[t], acc[t]);
        a = an;
#pragma unroll
        for (int t = 0; t < 4; ++t) bf[t] = bn[t];
        Xk = Xn;
    }
#pragma unroll
    for (int t = 0; t < 4; ++t) acc[t] = wmma4(a, bf[t], acc[t]);

#pragma unroll
    for (int v = 0; v < 8; ++v) {
        const int row = mbase + v + 8 * hi;
        const float bia = bias[row];
        float sc = 1.f, sh = 0.f;
        if (do_bn) {
            const float inv = rsqrtf(bnv[row] + 1e-5f);
            sc = bng[row] * inv;
            sh = bnb[row] - bnm[row] * sc;
        }
#pragma unroll
        for (int t = 0; t < 4; ++t) {
            const int col = nb0 + t * 16 + li;
            float val = acc[t][v] + bia;
            if (do_bn) val = val * sc + sh;
            Yb[(long)row * N + col] = val;
        }
    }
}

// ---------------------------------------------------------------------------
// Depthwise 9x9 stride-2 conv (pad 4): q[B,C,32,32] -> off[B,C,16,16]
// ---------------------------------------------------------------------------
__global__ __launch_bounds__(256) void dwconv_kernel(
    const float* __restrict__ q, const float* __restrict__ wdw,
    const float* __restrict__ bdw, float* __restrict__ off)
{
    const int idx = blockIdx.x * blockDim.x + threadIdx.x;  // (b*C+c)*256+s
    if (idx >= B_ * C_ * NS_) return;
    const int s  = idx & 255;
    const int bc = idx >> 8;
    const int c  = bc & (C_ - 1);
    const int oy = s >> 4, ox = s & 15;
    const float* qp = q + (long)bc * HW_;
    const float* wp = wdw + c * 81;
    float acc = bdw[c];
#pragma unroll
    for (int ky = 0; ky < 9; ++ky) {
        const int y = oy * 2 - 4 + ky;
        if ((unsigned)y >= 32u) continue;
#pragma unroll
        for (int kx = 0; kx < 9; ++kx) {
            const int x = ox * 2 - 4 + kx;
            if ((unsigned)x >= 32u) continue;
            acc = fmaf(qp[y * 32 + x], wp[ky * 9 + kx], acc);
        }
    }
    off[idx] = acc;
}

// ---------------------------------------------------------------------------
// Per (b, s): channel LN -> GELU -> 1x1 conv to 2 -> tanh(+ref) -> bilinear
// taps (4 weights + 4 indices) for grid_sample(align_corners=True, zeros pad).
// ---------------------------------------------------------------------------
__device__ __forceinline__ float block_reduce_sum(float v, float* red, int c) {
    red[c] = v; __syncthreads();
#pragma unroll
    for (int st = 128; st > 0; st >>= 1) {
        if (c < st) red[c] += red[c + st];
        __syncthreads();
    }
    const float r = red[0];
    __syncthreads();
    return r;
}

__global__ __launch_bounds__(256) void offset_kernel(
    const float* __restrict__ off, const float* __restrict__ lng,
    const float* __restrict__ lnb, const float* __restrict__ pw,
    float* __restrict__ sampw, int* __restrict__ sampi)
{
    __shared__ float red[256];
    const int bs = blockIdx.x;
    const int b = bs >> 8, s = bs & 255;
    const int c = threadIdx.x;

    const float val  = off[((long)(b * C_ + c)) * NS_ + s];
    const float mean = block_reduce_sum(val, red, c) * (1.f / 256.f);
    const float d    = val - mean;
    const float var  = block_reduce_sum(d * d, red, c) * (1.f / 256.f);
    const float xn   = d * rsqrtf(var + 1e-5f) * lng[c] + lnb[c];
    const float ge   = 0.5f * xn * (1.f + erff(xn * 0.70710678118654752f));
    const float oy_  = block_reduce_sum(ge * pw[c], red, c);
    const float ox_  = block_reduce_sum(ge * pw[C_ + c], red, c);

    if (c == 0) {
        const float ry = ((float)(s >> 4) + 0.5f) * (2.f / 16.f) - 1.f;
        const float rx = ((float)(s & 15) + 0.5f) * (2.f / 16.f) - 1.f;
        const float py = tanhf(oy_ + ry);
        const float px = tanhf(ox_ + rx);
        const float gx = (px + 1.f) * 0.5f * 31.f;   // align_corners=True
        const float gy = (py + 1.f) * 0.5f * 31.f;
        const float x0 = floorf(gx), y0 = floorf(gy);
        const float wx1 = gx - x0, wx0 = 1.f - wx1;
        const float wy1 = gy - y0, wy0 = 1.f - wy1;
        const int xi = (int)x0, yi = (int)y0;
        const float cw[4] = { wy0 * wx0, wy0 * wx1, wy1 * wx0, wy1 * wx1 };
        const int  cy[4] = { yi, yi, yi + 1, yi + 1 };
        const int  cx[4] = { xi, xi + 1, xi, xi + 1 };
        const int out = (b * NS_ + s) * 4;
#pragma unroll
        for (int j = 0; j < 4; ++j) {
            const bool valid = (cy[j] >= 0) && (cy[j] < 32) &&
                               (cx[j] >= 0) && (cx[j] < 32);
            const int yy = min(max(cy[j], 0), 31);
            const int xx = min(max(cx[j], 0), 31);
            sampw[out + j] = valid ? cw[j] : 0.f;
            sampi[out + j] = yy * 32 + xx;
        }
    }
}

// ---------------------------------------------------------------------------
// Bilinear gather: xs[b,c,s] = sum_j w_j * x[b,c,idx_j]
// ---------------------------------------------------------------------------
__global__ __launch_bounds__(256) void sample_kernel(
    const float* __restrict__ x, const float* __restrict__ sampw,
    const int* __restrict__ sampi, float* __restrict__ xs)
{
    const int idx = blockIdx.x * blockDim.x + threadIdx.x;
    if (idx >= B_ * C_ * NS_) return;
    const int s  = idx & 255;
    const int bc = idx >> 8;
    const int b  = bc >> 8;
    const float4 w4 = ((const float4*)sampw)[b * NS_ + s];
    const int4   i4 = ((const int4*)sampi)[b * NS_ + s];
    const float* xp = x + (long)bc * HW_;
    xs[idx] = w4.x * xp[i4.x] + w4.y * xp[i4.y] +
              w4.z * xp[i4.z] + w4.w * xp[i4.w];
}

// ---------------------------------------------------------------------------
// Fused attention per (b, head): S = (q_h)^T k_h * 1/4, row softmax over 256,
// out = P * V^T. One wave = 16 query rows; 4 waves / block; P staged in LDS.
// ---------------------------------------------------------------------------
__global__ __launch_bounds__(128) void attn_kernel(
    const float* __restrict__ q, const float* __restrict__ kmat,
    const float* __restrict__ vmat, float* __restrict__ outbuf)
{
    extern __shared__ float Pl[];   // 4 waves * 16*256 floats = 64 KB
    const int lane = threadIdx.x & 31;
    const int wid  = threadIdx.x >> 5;
    const int li   = lane & 15;
    const int hi   = lane >> 4;
    const int b = blockIdx.x >> 4, h = blockIdx.x & 15;
    const int mbase = blockIdx.y * 64 + wid * 16;

    const long head = (long)b * C_ + h * 16;
    const float* qh = q    + head * HW_;
    const float* kh = kmat + head * NS_;
    const float* vh = vmat + head * NS_;

    // A fragments for S (K = DH = 16): A[m,kk] = q_h[kk, m]
    v2f afr[4];
    const int m = mbase + li;
#pragma unroll
    for (int kc = 0; kc < 4; ++kc) {
        const int kk = kc * 4 + hi * 2;
        afr[kc].x = qh[(long)kk * HW_ + m];
        afr[kc].y = qh[(long)(kk + 1) * HW_ + m];
    }

    // Double-buffered B fragments across the 16 score tiles.
    const float* kcol = kh + li;
    v2f bcur[4], bnxt[4];
#pragma unroll
    for (int kc = 0; kc < 4; ++kc) {
        const int kk = kc * 4 + hi * 2;
        bcur[kc].x = kcol[(long)kk * NS_];
        bcur[kc].y = kcol[(long)(kk + 1) * NS_];
    }

    v8f acc[16];
#pragma unroll
    for (int nt = 0; nt < 16; ++nt) {
        if (nt < 15) {
            const float* kc2 = kcol + (nt + 1) * 16;
#pragma unroll
            for (int kc = 0; kc < 4; ++kc) {
                const int kk = kc * 4 + hi * 2;
                bnxt[kc].x = kc2[(long)kk * NS_];
                bnxt[kc].y = kc2[(long)(kk + 1) * NS_];
            }
        }
        v8f a_ = (v8f){0.f,0.f,0.f,0.f,0.f,0.f,0.f,0.f};
#pragma unroll
        for (int kc = 0; kc < 4; ++kc) a_ = wmma4(afr[kc], bcur[kc], a_);
        acc[nt] = a_;
#pragma unroll
        for (int kc = 0; kc < 4; ++kc) bcur[kc] = bnxt[kc];
    }

    // Row softmax. Row r = v + 8*hi; its 256 values live across the 16 lanes
    // of this half-wave (one per tile). Shuffle masks 1/2/4/8 stay in-half.
    const float scale = 0.25f;   // DH^-0.5
#pragma unroll
    for (int v = 0; v < 8; ++v) {
        float mx = -3.4e38f;
#pragma unroll
        for (int nt = 0; nt < 16; ++nt) mx = fmaxf(mx, acc[nt][v]);
        mx = fmaxf(mx, __shfl_xor(mx, 1));
        mx = fmaxf(mx, __shfl_xor(mx, 2));
        mx = fmaxf(mx, __shfl_xor(mx, 4));
        mx = fmaxf(mx, __shfl_xor(mx, 8));
        float sum = 0.f;
#pragma unroll
        for (int nt = 0; nt < 16; ++nt) {
            const float e = __expf((acc[nt][v] - mx) * scale);
            acc[nt][v] = e;
            sum += e;
        }
        sum += __shfl_xor(sum, 1);
        sum += __shfl_xor(sum, 2);
        sum += __shfl_xor(sum, 4);
        sum += __shfl_xor(sum, 8);
        const float rinv = 1.f / sum;
#pragma unroll
        for (int nt = 0; nt < 16; ++nt) acc[nt][v] *= rinv;
    }

    // Stage P (16x256, row-major) in LDS to re-shape into A-fragment order.
    float* Pw = Pl + wid * (16 * 256);
#pragma unroll
    for (int nt = 0; nt < 16; ++nt)
#pragma unroll
        for (int v = 0; v < 8; ++v)
            Pw[(v + 8 * hi) * 256 + nt * 16 + li] = acc[nt][v];
    __syncthreads();

    // out[m, dh] = sum_n P[m,n] * v_h[dh, n]   (double-buffered LDS + global)
    v8f oacc = (v8f){0.f,0.f,0.f,0.f,0.f,0.f,0.f,0.f};
    const float* Pr = Pw + li * 256 + hi * 2;
    const float* vr = vh + (long)li * NS_ + hi * 2;
    v2f pa = *(const v2f*)(Pr);
    v2f va = *(const v2f*)(vr);
    for (int k = 0; k < 252; k += 4) {
        v2f pn = *(const v2f*)(Pr + k + 4);
        v2f vn = *(const v2f*)(vr + k + 4);
        oacc = wmma4(pa, va, oacc);
        pa = pn; va = vn;
    }
    oacc = wmma4(pa, va, oacc);

    // D tile: row = v + 8*hi (query m), col = li (dh). Each lane stores 8
    // consecutive m values for its channel -> two aligned float4 stores.
    float* op = outbuf + head * HW_ + (long)li * HW_ + mbase + 8 * hi;
    float4 lo4 = { oacc[0], oacc[1], oacc[2], oacc[3] };
    float4 hi4 = { oacc[4], oacc[5], oacc[6], oacc[7] };
    *(float4*)op       = lo4;
    *(float4*)(op + 4) = hi4;
}

// ---------------------------------------------------------------------------
extern "C" void kernel_launch(void* const* d_in, const int* in_sizes, int n_in,
                              void* d_out, int out_size, void* d_ws, size_t ws_size,
                              hipStream_t stream)
{
    (void)in_sizes; (void)n_in; (void)out_size; (void)ws_size;
    const float* x    = (const float*)d_in[0];
    const float* wq   = (const float*)d_in[1];
    const float* bq   = (const float*)d_in[2];
    const float* dww  = (const float*)d_in[3];
    const float* dwb  = (const float*)d_in[4];
    const float* lng  = (const float*)d_in[5];
    const float* lnb  = (const float*)d_in[6];
    const float* pww  = (const float*)d_in[7];
    const float* wk   = (const float*)d_in[8];
    const float* bk   = (const float*)d_in[9];
    const float* wv   = (const float*)d_in[10];
    const float* bv   = (const float*)d_in[11];
    const float* wo   = (const float*)d_in[12];
    const float* bo   = (const float*)d_in[13];
    const float* bng  = (const float*)d_in[14];
    const float* bnb  = (const float*)d_in[15];
    const float* bnm  = (const float*)d_in[16];
    const float* bnv  = (const float*)d_in[17];
    float* out = (float*)d_out;

    float* qbuf  = (float*)d_ws;                               // B*C*HW
    float* offb  = qbuf  + (size_t)B_ * C_ * HW_;              // B*C*NS
    float* xsb   = offb  + (size_t)B_ * C_ * NS_;              // B*C*NS
    float* kb    = xsb   + (size_t)B_ * C_ * NS_;              // B*C*NS
    float* vb    = kb    + (size_t)B_ * C_ * NS_;              // B*C*NS
    float* aout  = vb    + (size_t)B_ * C_ * NS_;              // B*C*HW
    float* sampw = aout  + (size_t)B_ * C_ * HW_;              // B*NS*4
    int*   sampi = (int*)(sampw + (size_t)B_ * NS_ * 4);       // B*NS*4

    const long sBig = (long)C_ * HW_;   // per-batch stride for [C,HW] tensors
    const long sSml = (long)C_ * NS_;   // per-batch stride for [C,NS] tensors

    // 1) q = wq * x + bq               [B,256,1024]
    gemm_bias_kernel<<<dim3(B_, 16, 4), 128, 0, stream>>>(
        wq, x, bq, qbuf, HW_, C_, sBig, sBig,
        nullptr, nullptr, nullptr, nullptr, 0);

    // 2) depthwise 9x9 s2 conv         [B,256,16,16]
    dwconv_kernel<<<(B_ * C_ * NS_) / 256, 256, 0, stream>>>(qbuf, dww, dwb, offb);

    // 3) LN + GELU + pw conv + tanh + bilinear tap precompute
    offset_kernel<<<B_ * NS_, 256, 0, stream>>>(offb, lng, lnb, pww, sampw, sampi);

    // 4) deformable sampling           xs [B,256,256]
    sample_kernel<<<(B_ * C_ * NS_) / 256, 256, 0, stream>>>(x, sampw, sampi, xsb);

    // 5) k, v projections              [B,256,256]
    gemm_bias_kernel<<<dim3(B_, 16, 1), 128, 0, stream>>>(
        wk, xsb, bk, kb, NS_, C_, sSml, sSml,
        nullptr, nullptr, nullptr, nullptr, 0);
    gemm_bias_kernel<<<dim3(B_, 16, 1), 128, 0, stream>>>(
        wv, xsb, bv, vb, NS_, C_, sSml, sSml,
        nullptr, nullptr, nullptr, nullptr, 0);

    // 6) fused attention               aout [B,256,1024]
    attn_kernel<<<dim3(B_ * NH_, 16), 128, 4 * 16 * 256 * sizeof(float), stream>>>(
        qbuf, kb, vb, aout);

    // 7) output projection + BatchNorm (eval)
    gemm_bias_kernel<<<dim3(B_, 16, 4), 128, 0, stream>>>(
        wo, aout, bo, out, HW_, C_, sBig, sBig,
        bng, bnb, bnm, bnv, 1);
}